// PRNN_1769526526374
// MI455X (gfx1250) — compile-verified
//
#include <hip/hip_runtime.h>

typedef float v2f __attribute__((ext_vector_type(2)));
typedef float v8f __attribute__((ext_vector_type(8)));

#define B_TOT   4096
#define S_TOT   512
#define MPTS    128
#define N_LAT   384
#define CHUNK_T 16
#define NCHUNK  (S_TOT / CHUNK_T)     // 32
#define LAT_STRIDE 388                // pad 384 -> 388 to avoid LDS bank conflicts on A reads

// Material constants (computed in double, used as f32 like the reference)
constexpr double E_d = 3130.0, NU_d = 0.3;
constexpr float MU_F     = (float)(E_d / (2.0 * (1.0 + NU_d)));                       // 1203.846
constexpr float LAM_F    = (float)(E_d * NU_d / ((1.0 + NU_d) * (1.0 - 2.0 * NU_d))); // 1805.769
constexpr float TWO_MU   = 2.0f * MU_F;
constexpr float THREE_MU = 3.0f * MU_F;
constexpr float A_HF     = 64.8f;
constexpr float B_HF     = 33.6f;
constexpr float INV_C    = (float)(1.0 / 0.003407);
constexpr float B_OVER_C = (float)(33.6 / 0.003407);

__global__ __launch_bounds__(MPTS) void prnn_fused_kernel(
    const float* __restrict__ x,    // [4096, 512, 3]
    const float* __restrict__ W1,   // [384, 3]
    const float* __restrict__ W2,   // [3, 384]
    float* __restrict__ out)        // [4096, 512, 3]
{
    __shared__ float lat_lds[CHUNK_T * LAT_STRIDE];   // 16 timesteps x 384 latents (padded)
    __shared__ float w2_lds[4 * N_LAT];               // W2 staged + one zero row for dead cols
    __shared__ float red_lds[4 * 16 * 16];            // per-wave partial C tiles
    __shared__ float xs_lds[CHUNK_T * 3];             // x chunk

    const int tid = threadIdx.x;
    const int b   = blockIdx.x;
    const int m   = tid;                              // material point 0..127

    // Stage W2 into LDS; row 3 is all zeros (mapped to by the 13 dead output cols)
    for (int i = tid; i < 4 * N_LAT; i += MPTS)
        w2_lds[i] = (i < 3 * N_LAT) ? W2[i] : 0.0f;

    // Per-thread 3x3 W1 block: rows 3m..3m+2
    float w[9];
#pragma unroll
    for (int i = 0; i < 9; ++i) w[i] = W1[m * 9 + i];

    // J2 state in registers for the whole sequence
    float ep0 = 0.0f, ep1 = 0.0f, ep2 = 0.0f, ep3 = 0.0f, kappa = 0.0f;

    const size_t xrow = (size_t)b * S_TOT * 3;

    // WMMA lane decomposition (wave32)
    const int wave  = tid >> 5;
    const int lane  = tid & 31;
    const int arow  = lane & 15;            // A-matrix row (timestep) / B-matrix col (output)
    const int brow  = (arow < 3) ? arow : 3;// dead cols read the zero row -> branch-free B load
    const int khalf = (lane >> 4) << 1;     // K sub-offset 0 or 2
    const int kbase = wave * 96;            // each wave handles 96 of K=384

    // Hoisted LDS base pointers for the WMMA fragment loads (8B-aligned: K is even)
    const float* aptr = &lat_lds[arow * LAT_STRIDE + kbase + khalf];
    const float* bptr = &w2_lds[brow * N_LAT + kbase + khalf];

    for (int chunk = 0; chunk < NCHUNK; ++chunk) {
        const int t0 = chunk * CHUNK_T;

        __syncthreads();  // prev chunk fully consumed
        if (tid < CHUNK_T * 3) xs_lds[tid] = x[xrow + (size_t)t0 * 3 + tid];
        if (tid == 0 && chunk + 1 < NCHUNK)
            __builtin_prefetch(&x[xrow + (size_t)(t0 + CHUNK_T) * 3], 0, 1);
        __syncthreads();  // xs ready

        // ---- sequential J2 scan over 16 timesteps ----
#pragma unroll 1
        for (int tt = 0; tt < CHUNK_T; ++tt) {
            const float xs0 = xs_lds[tt * 3 + 0];
            const float xs1 = xs_lds[tt * 3 + 1];
            const float xs2 = xs_lds[tt * 3 + 2];
            // fc1: per-point 3-vector of strains
            const float eps0 = w[0] * xs0 + w[1] * xs1 + w[2] * xs2;
            const float eps1 = w[3] * xs0 + w[4] * xs1 + w[5] * xs2;
            const float eps2 = w[6] * xs0 + w[7] * xs1 + w[8] * xs2;
            // elastic trial
            const float ee0 = eps0 - ep0;
            const float ee1 = eps1 - ep1;
            const float ee2 = -ep2;
            const float ee3 = 0.5f * eps2 - ep3;
            const float lt  = LAM_F * (ee0 + ee1 + ee2);
            const float sig0 = TWO_MU * ee0 + lt;
            const float sig1 = TWO_MU * ee1 + lt;
            const float sig2 = TWO_MU * ee2 + lt;
            const float sig3 = TWO_MU * ee3;
            const float p  = (sig0 + sig1 + sig2) * (1.0f / 3.0f);
            const float s0 = sig0 - p, s1 = sig1 - p, s2 = sig2 - p, s3 = sig3;
            const float q  = sqrtf(1.5f * (s0 * s0 + s1 * s1 + s2 * s2 + 2.0f * s3 * s3));
            const float f  = q - (A_HF - B_HF * __expf(-kappa * INV_C));

            float dg = 0.0f;
            if (f > 0.0f) {   // identical to the reference's where(f>0, dg, 0)
#pragma unroll
                for (int it = 0; it < 12; ++it) {
                    const float ex = __expf(-(kappa + dg) * INV_C);
                    const float r  = q - THREE_MU * dg - (A_HF - B_HF * ex);
                    const float dr = -THREE_MU - B_OVER_C * ex;
                    dg = fmaxf(dg - r / dr, 0.0f);
                }
            }
            const float qs   = fmaxf(q, 1e-12f);
            const float coef = 1.5f * dg / qs;       // dg * n = coef * s
            ep0 += coef * s0;  ep1 += coef * s1;
            ep2 += coef * s2;  ep3 += coef * s3;
            kappa += dg;
            const float tmc = TWO_MU * coef;
            float* lrow = &lat_lds[tt * LAT_STRIDE + m * 3];
            lrow[0] = sig0 - tmc * s0;
            lrow[1] = sig1 - tmc * s1;
            lrow[2] = sig3 - tmc * s3;
        }
        __syncthreads();  // lat chunk ready

        // ---- fc2 via WMMA: [16 x 384] · [384 x 16(3 live)] , K-slice per wave ----
        v8f acc = {};
#pragma unroll 4
        for (int kb = 0; kb < 96; kb += 4) {
            const v2f a  = *(const v2f*)(aptr + kb);
            const v2f bv = *(const v2f*)(bptr + kb);
            acc = __builtin_amdgcn_wmma_f32_16x16x4_f32(
                false, a, false, bv, (short)0, acc, false, false);
        }
        // store per-wave partial 16x16 tile
#pragma unroll
        for (int i = 0; i < 8; ++i) {
            const int r = i + ((lane >> 4) << 3);   // VGPR i: M=i (lanes 0-15), M=8+i (lanes 16-31)
            red_lds[wave * 256 + r * 16 + (lane & 15)] = acc[i];
        }
        __syncthreads();

        // reduce 4 wave partials and emit 16 timesteps x 3 outputs
        if (tid < CHUNK_T * 3) {
            const int r = tid / 3, n = tid - r * 3;
            const int off = r * 16 + n;
            const float v = red_lds[off] + red_lds[256 + off] +
                            red_lds[512 + off] + red_lds[768 + off];
            out[((size_t)b * S_TOT + t0 + r) * 3 + n] = v;
        }
    }
}

extern "C" void kernel_launch(void* const* d_in, const int* in_sizes, int n_in,
                              void* d_out, int out_size, void* d_ws, size_t ws_size,
                              hipStream_t stream) {
    (void)in_sizes; (void)n_in; (void)out_size; (void)d_ws; (void)ws_size;
    const float* x  = (const float*)d_in[0];
    const float* W1 = (const float*)d_in[1];
    const float* W2 = (const float*)d_in[2];
    float* out = (float*)d_out;
    hipLaunchKernelGGL(prnn_fused_kernel, dim3(B_TOT), dim3(MPTS), 0, stream,
                       x, W1, W2, out);
}